// MaEAALayer_8143257994025
// MI455X (gfx1250) — compile-verified
//
#include <hip/hip_runtime.h>
#include <hip/hip_bf16.h>
#include <math.h>

// ---------------------------------------------------------------------------
// Problem constants (match the JAX reference)
// ---------------------------------------------------------------------------
constexpr int BATCH   = 8;
constexpr int SEQ     = 2048;
constexpr int DMODEL  = 1024;
constexpr int DINNER  = 2048;   // EXPAND * DMODEL
constexpr int DSTATE  = 16;
constexpr int DTRANK  = 64;
constexpr int NTOK    = BATCH * SEQ;          // 16384 rows in every GEMM

// WMMA GEMM tiling: 128x64 block tile, 8 wave32s in a 4x2 grid, 32x32 per wave
constexpr int BLK_M = 128;
constexpr int BLK_N = 64;
constexpr int BLK_K = 16;
constexpr int LDSK  = BLK_K + 4;    // row stride 20 floats: bank-conflict-free

typedef __attribute__((ext_vector_type(2))) float v2f;
typedef __attribute__((ext_vector_type(8))) float v8f;

// ---------------------------------------------------------------------------
// device helpers
// ---------------------------------------------------------------------------
__device__ __forceinline__ float gelu_exact(float x) {
    return 0.5f * x * (1.0f + erff(x * 0.70710678118654752f));
}
__device__ __forceinline__ float softplus_f(float x) {
    return (x > 20.0f) ? x : log1pf(__expf(x));
}
__device__ __forceinline__ float silu_f(float x) {
    return x / (1.0f + __expf(-x));
}

// CDNA5 async global->LDS copy (ASYNCcnt-tracked, no VGPR round trip).
// lds_off: LDS byte address (low 32 bits of a generic pointer to __shared__).
__device__ __forceinline__ void async_load_b128(unsigned lds_off, const float* gaddr) {
    asm volatile("global_load_async_to_lds_b128 %0, %1, off"
                 :: "v"(lds_off), "v"(gaddr) : "memory");
}
__device__ __forceinline__ void wait_asynccnt0() {
    asm volatile("s_wait_asynccnt 0x0" ::: "memory");
}

// blockDim.x must be 256
__device__ __forceinline__ float block_reduce_sum(float v, float* red) {
    const int tid = threadIdx.x;
    red[tid] = v;
    __syncthreads();
    #pragma unroll
    for (int off = 128; off > 0; off >>= 1) {
        if (tid < off) red[tid] += red[tid + off];
        __syncthreads();
    }
    float r = red[0];
    __syncthreads();
    return r;
}

// ---------------------------------------------------------------------------
// fp32 WMMA GEMM:  C[M,N] = act( A[M,K] @ W[N,K]^T + bias )
//   (torch Linear layout: weight rows are output features)
// ACT: 0 = none, 1 = exact gelu, 2 = softplus
// GUARD: bounds-checked staging/epilogue (only needed when N % 64 != 0)
// ---------------------------------------------------------------------------
template <int ACT, bool HAS_BIAS, bool GUARD>
__global__ __launch_bounds__(256)
void gemm_f32_wmma(const float* __restrict__ A, int lda,
                   const float* __restrict__ W, int ldw,
                   const float* __restrict__ bias,
                   float* __restrict__ C, int ldc,
                   int M, int N, int K)
{
    __shared__ float As[BLK_M][LDSK];   // [m][k]
    __shared__ float Bs[BLK_N][LDSK];   // [n][k]

    const int m0   = blockIdx.y * BLK_M;
    const int n0   = blockIdx.x * BLK_N;
    const int tid  = threadIdx.x;
    const int wid  = tid >> 5;          // 8 wave32s
    const int lane = tid & 31;
    const int half = lane >> 4;         // 0: lanes 0-15, 1: lanes 16-31
    const int lm   = lane & 15;
    const int am0  = (wid & 3) * 32;    // wave M offset inside tile
    const int bn0  = (wid >> 2) * 32;   // wave N offset inside tile

    v8f acc00 = {}, acc01 = {}, acc10 = {}, acc11 = {};

    // per-thread staging coordinates (constant across the k-loop)
    const int srow = tid >> 2;          // 0..63
    const int skq  = (tid & 3) * 4;     // 0,4,8,12
    const unsigned ldsA0 = (unsigned)(size_t)&As[srow][skq];
    const unsigned ldsA1 = (unsigned)(size_t)&As[srow + 64][skq];
    const unsigned ldsB  = (unsigned)(size_t)&Bs[srow][skq];
    const float* gA0 = A + (size_t)(m0 + srow)      * lda + skq;
    const float* gA1 = A + (size_t)(m0 + srow + 64) * lda + skq;
    const float* gB  = W + (size_t)(n0 + srow)      * ldw + skq;

    for (int k0 = 0; k0 < K; k0 += BLK_K) {
        if (!GUARD) {
            // ---- async staging: global -> LDS, tracked by ASYNCcnt --------
            async_load_b128(ldsA0, gA0 + k0);
            async_load_b128(ldsA1, gA1 + k0);
            async_load_b128(ldsB,  gB  + k0);
            // speculative prefetch of the next k-tile (global_prefetch_b8)
            if (k0 + BLK_K < K) {
                __builtin_prefetch(&A[(size_t)(m0 + (tid >> 1)) * lda + k0 + BLK_K], 0, 0);
                __builtin_prefetch(&W[(size_t)(n0 + (tid >> 2)) * ldw + k0 + BLK_K], 0, 0);
            }
            wait_asynccnt0();
        } else {
            // ---- guarded scalar staging -----------------------------------
            #pragma unroll
            for (int i = 0; i < (BLK_M * BLK_K) / 256; ++i) {
                const int e  = tid + i * 256;
                const int m  = e >> 4;
                const int kk = e & 15;
                const int gm = m0 + m;
                As[m][kk] = (gm < M) ? A[(size_t)gm * lda + k0 + kk] : 0.0f;
            }
            #pragma unroll
            for (int i = 0; i < (BLK_N * BLK_K) / 256; ++i) {
                const int e  = tid + i * 256;
                const int n  = e >> 4;
                const int kk = e & 15;
                const int gn = n0 + n;
                Bs[n][kk] = (gn < N) ? W[(size_t)gn * ldw + k0 + kk] : 0.0f;
            }
        }
        __syncthreads();

        // ---- 4 WMMA k-steps of 4: contiguous ds_load_b64 fragments --------
        #pragma unroll
        for (int kk = 0; kk < BLK_K; kk += 4) {
            const int ka = kk + 2 * half;   // lanes 0-15: K=kk,kk+1 ; 16-31: kk+2,kk+3
            v2f a0, a1, b0, b1;
            a0.x = As[am0 + lm][ka];        a0.y = As[am0 + lm][ka + 1];
            a1.x = As[am0 + 16 + lm][ka];   a1.y = As[am0 + 16 + lm][ka + 1];
            b0.x = Bs[bn0 + lm][ka];        b0.y = Bs[bn0 + lm][ka + 1];
            b1.x = Bs[bn0 + 16 + lm][ka];   b1.y = Bs[bn0 + 16 + lm][ka + 1];
            acc00 = __builtin_amdgcn_wmma_f32_16x16x4_f32(false, a0, false, b0,
                                                          (short)0, acc00, false, false);
            acc01 = __builtin_amdgcn_wmma_f32_16x16x4_f32(false, a0, false, b1,
                                                          (short)0, acc01, false, false);
            acc10 = __builtin_amdgcn_wmma_f32_16x16x4_f32(false, a1, false, b0,
                                                          (short)0, acc10, false, false);
            acc11 = __builtin_amdgcn_wmma_f32_16x16x4_f32(false, a1, false, b1,
                                                          (short)0, acc11, false, false);
        }
        __syncthreads();
    }

    // ---- epilogue: lane holds rows vr (+8 for lanes 16-31), col lm ---------
    #pragma unroll
    for (int fm = 0; fm < 2; ++fm) {
        #pragma unroll
        for (int vr = 0; vr < 8; ++vr) {
            const int gm = m0 + am0 + fm * 16 + vr + (half ? 8 : 0);
            if (GUARD && gm >= M) continue;
            #pragma unroll
            for (int fn = 0; fn < 2; ++fn) {
                const int gn = n0 + bn0 + fn * 16 + lm;
                if (GUARD && gn >= N) continue;
                const v8f& acc = fm == 0 ? (fn == 0 ? acc00 : acc01)
                                         : (fn == 0 ? acc10 : acc11);
                float v = acc[vr];
                if (HAS_BIAS) v += bias[gn];
                if (ACT == 1) v = gelu_exact(v);
                else if (ACT == 2) v = softplus_f(v);
                C[(size_t)gm * ldc + gn] = v;
            }
        }
    }
}

// ---------------------------------------------------------------------------
// depthwise causal conv1d (taps=4) + SiLU:  reads u = XZ[:, :, 0:2048]
// ---------------------------------------------------------------------------
__global__ __launch_bounds__(256)
void conv_silu_kernel(const float* __restrict__ xz, const float* __restrict__ cw,
                      const float* __restrict__ cb, float* __restrict__ uc)
{
    const size_t i = (size_t)blockIdx.x * 256 + threadIdx.x;   // over NTOK*DINNER
    const int d  = (int)(i & (DINNER - 1));
    const int bl = (int)(i >> 11);
    const int l  = bl & (SEQ - 1);
    const int b  = bl >> 11;
    float acc = cb[d];
    #pragma unroll
    for (int k = 0; k < 4; ++k) {
        const int li = l + k - 3;                              // causal: w[3] is current tap
        if (li >= 0)
            acc += xz[((size_t)(b * SEQ + li) << 12) + d] * cw[d * 4 + k];
    }
    uc[i] = silu_f(acc);
}

// ---------------------------------------------------------------------------
// selective scan: one thread per (b, d) channel, 16-state recurrence in regs.
// Fuses the D_skip and silu(res) gating epilogue.
// ---------------------------------------------------------------------------
__global__ __launch_bounds__(256)
void scan_kernel(const float* __restrict__ delta, const float* __restrict__ uc,
                 const float* __restrict__ xdbl,  const float* __restrict__ xz,
                 const float* __restrict__ A_log, const float* __restrict__ D_skip,
                 float* __restrict__ y)
{
    const int t = blockIdx.x * 256 + threadIdx.x;   // 16384 channels
    const int b = t >> 11;
    const int d = t & (DINNER - 1);

    float An[DSTATE], h[DSTATE];
    #pragma unroll
    for (int n = 0; n < DSTATE; ++n) {
        An[n] = -__expf(A_log[d * DSTATE + n]);
        h[n]  = 0.0f;
    }
    const float dsk = D_skip[d];

    for (int l = 0; l < SEQ; ++l) {
        const size_t row = (size_t)(b * SEQ + l);
        const float dt = delta[(row << 11) + d];
        const float u  = uc[(row << 11) + d];
        const float* xd = xdbl + row * 96;          // [dt(64) | B(16) | C(16)]
        float acc = 0.0f;
        #pragma unroll
        for (int n = 0; n < DSTATE; ++n) {
            const float dA = __expf(dt * An[n]);
            h[n] = dA * h[n] + (dt * xd[DTRANK + n]) * u;
            acc += h[n] * xd[DTRANK + DSTATE + n];
        }
        const float res = xz[(row << 12) + DINNER + d];
        y[(row << 11) + d] = (acc + u * dsk) * silu_f(res);
    }
}

// ---------------------------------------------------------------------------
// LayerNorm over D=1024 (optionally adds a residual first)
// ---------------------------------------------------------------------------
__global__ __launch_bounds__(256)
void layernorm_kernel(const float* __restrict__ x, const float* __restrict__ resid,
                      const float* __restrict__ g, const float* __restrict__ b,
                      float* __restrict__ out)
{
    __shared__ float red[256];
    const int row = blockIdx.x;
    const size_t base = (size_t)row * DMODEL;
    float v[4];
    float s = 0.0f;
    #pragma unroll
    for (int i = 0; i < 4; ++i) {
        const int c = threadIdx.x + i * 256;
        float t = x[base + c];
        if (resid) t += resid[base + c];
        v[i] = t;
        s += t;
    }
    const float mean = block_reduce_sum(s, red) * (1.0f / DMODEL);
    float s2 = 0.0f;
    #pragma unroll
    for (int i = 0; i < 4; ++i) { const float dd = v[i] - mean; s2 += dd * dd; }
    const float var = block_reduce_sum(s2, red) * (1.0f / DMODEL);
    const float inv = rsqrtf(var + 1e-12f);
    #pragma unroll
    for (int i = 0; i < 4; ++i) {
        const int c = threadIdx.x + i * 256;
        out[base + c] = (v[i] - mean) * inv * g[c] + b[c];
    }
}

// ---------------------------------------------------------------------------
// row-wise L2 norm; QMODE also applies exact gelu and emits qw = (q·w_g)/32
// ---------------------------------------------------------------------------
template <bool QMODE>
__global__ __launch_bounds__(256)
void l2norm_row_kernel(float* __restrict__ x, const float* __restrict__ wg,
                       float* __restrict__ qw)
{
    __shared__ float red[256];
    const int row = blockIdx.x;
    const size_t base = (size_t)row * DMODEL;
    float v[4];
    float s = 0.0f;
    #pragma unroll
    for (int i = 0; i < 4; ++i) {
        const int c = threadIdx.x + i * 256;
        const float t = x[base + c];
        v[i] = t;
        s += t * t;
    }
    const float n   = sqrtf(block_reduce_sum(s, red));
    const float inv = 1.0f / fmaxf(n, 1e-12f);
    float dot = 0.0f;
    #pragma unroll
    for (int i = 0; i < 4; ++i) {
        const int c = threadIdx.x + i * 256;
        float t = v[i] * inv;
        if (QMODE) { t = gelu_exact(t); dot += t * wg[c]; }
        x[base + c] = t;
    }
    if (QMODE) {
        const float qd = block_reduce_sum(dot, red);
        if (threadIdx.x == 0) qw[row] = qd * 0.03125f;   // D_MODEL^-0.5
    }
}

// per-batch L2 norm of qw over the sequence axis
__global__ __launch_bounds__(256)
void bnorm_kernel(const float* __restrict__ qw, float* __restrict__ bnrm)
{
    __shared__ float red[256];
    const int b = blockIdx.x;
    float s = 0.0f;
    for (int i = threadIdx.x; i < SEQ; i += 256) {
        const float t = qw[b * SEQ + i];
        s += t * t;
    }
    const float tot = block_reduce_sum(s, red);
    if (threadIdx.x == 0) bnrm[b] = sqrtf(tot);
}

// G[b,d] = sum_l (qw[b,l]/max(norm_b,eps)) * q[b,l,d]
__global__ __launch_bounds__(256)
void g_kernel(const float* __restrict__ q, const float* __restrict__ qw,
              const float* __restrict__ bnrm, float* __restrict__ G)
{
    const int t = blockIdx.x * 256 + threadIdx.x;   // BATCH*DMODEL = 8192
    const int b = t >> 10;
    const int d = t & (DMODEL - 1);
    const float inv = 1.0f / fmaxf(bnrm[b], 1e-12f);
    float s = 0.0f;
    for (int l = 0; l < SEQ; ++l)
        s += qw[(b << 11) + l] * q[((size_t)((b << 11) + l) << 10) + d];
    G[t] = s * inv;
}

// gk[b,l,d] = G[b,d] * k[b,l,d]
__global__ __launch_bounds__(256)
void gk_kernel(const float* __restrict__ G, const float* __restrict__ k,
               float* __restrict__ out)
{
    const size_t i = (size_t)blockIdx.x * 256 + threadIdx.x;
    const int d = (int)(i & (DMODEL - 1));
    const int b = (int)(i >> 21);                   // / (SEQ*DMODEL)
    out[i] = G[b * DMODEL + d] * k[i];
}

__global__ __launch_bounds__(256)
void mul_kernel(const float* __restrict__ a, const float* __restrict__ b,
                float* __restrict__ out)
{
    const size_t i = (size_t)blockIdx.x * 256 + threadIdx.x;
    out[i] = a[i] * b[i];
}

// ---------------------------------------------------------------------------
// host launcher
// ---------------------------------------------------------------------------
extern "C" void kernel_launch(void* const* d_in, const int* in_sizes, int n_in,
                              void* d_out, int out_size, void* d_ws, size_t ws_size,
                              hipStream_t stream)
{
    const float* x_in       = (const float*)d_in[0];
    const float* in_proj_w  = (const float*)d_in[1];
    const float* conv_w     = (const float*)d_in[2];
    const float* conv_b     = (const float*)d_in[3];
    const float* x_proj_w   = (const float*)d_in[4];
    const float* dt_proj_w  = (const float*)d_in[5];
    const float* dt_proj_b  = (const float*)d_in[6];
    const float* A_log      = (const float*)d_in[7];
    const float* D_skip     = (const float*)d_in[8];
    const float* out_proj_w = (const float*)d_in[9];
    const float* ln0_g = (const float*)d_in[10];
    const float* ln0_b = (const float*)d_in[11];
    const float* q_w   = (const float*)d_in[12];
    const float* q_b   = (const float*)d_in[13];
    const float* k_w   = (const float*)d_in[14];
    const float* k_b   = (const float*)d_in[15];
    const float* w_g   = (const float*)d_in[16];
    const float* proj_w = (const float*)d_in[17];
    const float* proj_b = (const float*)d_in[18];
    const float* fl_w  = (const float*)d_in[19];
    const float* fl_b  = (const float*)d_in[20];
    const float* lni_g = (const float*)d_in[21];
    const float* lni_b = (const float*)d_in[22];
    const float* ln1_g = (const float*)d_in[23];
    const float* ln1_b = (const float*)d_in[24];
    const float* d1_w  = (const float*)d_in[25];
    const float* d1_b  = (const float*)d_in[26];
    const float* d2_w  = (const float*)d_in[27];
    const float* d2_b  = (const float*)d_in[28];
    const float* ln2_g = (const float*)d_in[29];
    const float* ln2_b = (const float*)d_in[30];

    float* out = (float*)d_out;

    // workspace layout: 13 regions of NTOK*DMODEL floats (~840 MB), reused.
    const size_t R = (size_t)NTOK * DMODEL;
    float* ws   = (float*)d_ws;
    float* XZ   = ws;                 // regions 0-3 (16384 x 4096)   [mamba]
    float* UC   = ws + 4 * R;         // regions 4-5 (16384 x 2048)
    float* DEL  = ws + 6 * R;         // regions 6-7
    float* Yb   = ws + 8 * R;         // regions 8-9
    float* Mb   = ws + 10 * R;        // region 10
    float* Hb   = ws + 11 * R;        // region 11  (live through attention)
    float* XDBL = ws + 12 * R;        // region 12: 16384 x 96, plus smalls
    float* QW   = XDBL + (size_t)NTOK * 96;
    float* BNRM = QW + NTOK;
    float* Gb   = BNRM + 8;
    // attention-phase reuse (mamba temporaries are dead by then)
    float* Qb   = ws + 4 * R;
    float* Kb   = ws + 5 * R;
    float* GK   = ws + 6 * R;
    float* HS   = ws + 7 * R;
    float* LNI  = ws + 8 * R;
    float* HMUL = ws + 9 * R;
    float* HS2  = ws + 6 * R;         // reuse GK after proj GEMM consumed it
    float* H2   = ws + 10 * R;        // reuse Mb after h computed
    float* F1   = ws;                 // regions 0-3 (16384 x 4096)
    float* F2   = ws + 4 * R;         // reuse Qb

    const dim3 blk(256);
    auto ggrid = [](int M, int N) {
        return dim3((unsigned)((N + BLK_N - 1) / BLK_N),
                    (unsigned)((M + BLK_M - 1) / BLK_M));
    };

    // ---------------- Mamba ----------------
    // xz = x @ in_proj_w.T                                  [16384, 4096]
    gemm_f32_wmma<0, false, false><<<ggrid(NTOK, 2 * DINNER), blk, 0, stream>>>(
        x_in, DMODEL, in_proj_w, DMODEL, nullptr, XZ, 2 * DINNER, NTOK, 2 * DINNER, DMODEL);
    // u = silu(causal depthwise conv(u))                    [16384, 2048]
    conv_silu_kernel<<<(unsigned)(((size_t)NTOK * DINNER) / 256), blk, 0, stream>>>(
        XZ, conv_w, conv_b, UC);
    // x_dbl = u @ x_proj_w.T                                [16384, 96]  (N%64!=0 -> GUARD)
    gemm_f32_wmma<0, false, true><<<ggrid(NTOK, 96), blk, 0, stream>>>(
        UC, DINNER, x_proj_w, DINNER, nullptr, XDBL, 96, NTOK, 96, DINNER);
    // delta = softplus(x_dbl[:, :64] @ dt_proj_w.T + b)     [16384, 2048]
    gemm_f32_wmma<2, true, false><<<ggrid(NTOK, DINNER), blk, 0, stream>>>(
        XDBL, 96, dt_proj_w, DTRANK, dt_proj_b, DEL, DINNER, NTOK, DINNER, DTRANK);
    // selective scan (+ D_skip + silu(res) gate fused)      [16384, 2048]
    scan_kernel<<<NTOK / 256, blk, 0, stream>>>(DEL, UC, XDBL, XZ, A_log, D_skip, Yb);
    // m = y @ out_proj_w.T                                  [16384, 1024]
    gemm_f32_wmma<0, false, false><<<ggrid(NTOK, DMODEL), blk, 0, stream>>>(
        Yb, DINNER, out_proj_w, DINNER, nullptr, Mb, DMODEL, NTOK, DMODEL, DINNER);
    // h = LN(m + x)
    layernorm_kernel<<<NTOK, blk, 0, stream>>>(Mb, x_in, ln0_g, ln0_b, Hb);

    // ---------------- EAAttention ----------------
    gemm_f32_wmma<0, true, false><<<ggrid(NTOK, DMODEL), blk, 0, stream>>>(
        Hb, DMODEL, q_w, DMODEL, q_b, Qb, DMODEL, NTOK, DMODEL, DMODEL);
    gemm_f32_wmma<0, true, false><<<ggrid(NTOK, DMODEL), blk, 0, stream>>>(
        Hb, DMODEL, k_w, DMODEL, k_b, Kb, DMODEL, NTOK, DMODEL, DMODEL);
    // q = gelu(l2norm(q)); qw = (q @ w_g) * D^-0.5  (fused)
    l2norm_row_kernel<true><<<NTOK, blk, 0, stream>>>(Qb, w_g, QW);
    // k = l2norm(k)
    l2norm_row_kernel<false><<<NTOK, blk, 0, stream>>>(Kb, nullptr, nullptr);
    // Aw = l2norm(qw over seq); G = sum_l Aw*q
    bnorm_kernel<<<BATCH, blk, 0, stream>>>(QW, BNRM);
    g_kernel<<<(BATCH * DMODEL) / 256, blk, 0, stream>>>(Qb, QW, BNRM, Gb);
    // hs = (G ⊙ k) @ proj_w.T + proj_b
    gk_kernel<<<(unsigned)(R / 256), blk, 0, stream>>>(Gb, Kb, GK);
    gemm_f32_wmma<0, true, false><<<ggrid(NTOK, DMODEL), blk, 0, stream>>>(
        GK, DMODEL, proj_w, DMODEL, proj_b, HS, DMODEL, NTOK, DMODEL, DMODEL);
    // hs2 = (hs * LN(h)) @ fl_w.T + fl_b
    layernorm_kernel<<<NTOK, blk, 0, stream>>>(Hb, nullptr, lni_g, lni_b, LNI);
    mul_kernel<<<(unsigned)(R / 256), blk, 0, stream>>>(HS, LNI, HMUL);
    gemm_f32_wmma<0, true, false><<<ggrid(NTOK, DMODEL), blk, 0, stream>>>(
        HMUL, DMODEL, fl_w, DMODEL, fl_b, HS2, DMODEL, NTOK, DMODEL, DMODEL);
    // h2 = LN(hs2 + h)
    layernorm_kernel<<<NTOK, blk, 0, stream>>>(HS2, Hb, ln1_g, ln1_b, H2);

    // ---------------- FeedForward ----------------
    // f1 = gelu(h2 @ d1_w.T + d1_b)                         [16384, 4096]
    gemm_f32_wmma<1, true, false><<<ggrid(NTOK, 4 * DMODEL), blk, 0, stream>>>(
        H2, DMODEL, d1_w, DMODEL, d1_b, F1, 4 * DMODEL, NTOK, 4 * DMODEL, DMODEL);
    // f2 = f1 @ d2_w.T + d2_b                               [16384, 1024]
    gemm_f32_wmma<0, true, false><<<ggrid(NTOK, DMODEL), blk, 0, stream>>>(
        F1, 4 * DMODEL, d2_w, 4 * DMODEL, d2_b, F2, DMODEL, NTOK, DMODEL, 4 * DMODEL);
    // out = LN(f2 + h2)
    layernorm_kernel<<<NTOK, blk, 0, stream>>>(F2, H2, ln2_g, ln2_b, out);
}